// AdvancedUpSampling2D_15522011808059
// MI455X (gfx1250) — compile-verified
//
#include <hip/hip_runtime.h>

// Max-unpool scatter, bandwidth-roofline kernel for MI455X (gfx1250).
// updates: [16,64,64,256] f32, mask: same-shape int32 flat indices into
// out: [16,128,128,256] f32. Windows are disjoint -> one pass, no zero-init,
// no atomics: each thread owns one input float4 and writes its entire 2x2
// output window (selected slot = value, others = 0).
//
// Traffic: 128 MiB read + 256 MiB write = 384 MiB -> ~16.5 us floor @ 23.3 TB/s.

typedef __attribute__((ext_vector_type(4))) float v4f;  // native vector: OK for NT builtins
typedef __attribute__((ext_vector_type(4))) int   v4i;

namespace {
constexpr int B    = 16;
constexpr int H    = 64;
constexpr int W    = 64;
constexpr int C    = 256;
constexpr int Hout = H * 2;          // 128
constexpr int Wout = W * 2;          // 128
constexpr int C4   = C / 4;          // 64 vec4 per (b,y,x)
constexpr int TOTAL_T = B * H * W * C4;   // 4,194,304 threads
constexpr int BLOCK   = 256;              // 8 wave32 per block
} // namespace

__global__ __launch_bounds__(BLOCK) void
AdvancedUpSampling2D_unpool_kernel(const v4f* __restrict__ upd,
                                   const v4i* __restrict__ msk,
                                   v4f*       __restrict__ out)
{
    const int tid = blockIdx.x * BLOCK + threadIdx.x;

    // Prefetch the streams ~16KB ahead (global_prefetch_b8, speculative:
    // out-of-range at the tail is silently dropped, no counter traffic).
    __builtin_prefetch(reinterpret_cast<const char*>(upd + tid) + 16384, 0, 1);
    __builtin_prefetch(reinterpret_cast<const char*>(msk + tid) + 16384, 0, 1);

    // Read-once streams: non-temporal (TH=NT) B128 loads.
    const v4f v = __builtin_nontemporal_load(&upd[tid]);
    const v4i m = __builtin_nontemporal_load(&msk[tid]);

    // tid = ((b*H + h)*W + w)*C4 + c4  (matches [B,H,W,C] row-major with c
    // grouped by 4), so input loads are fully linear/coalesced.
    const int c4 = tid & (C4 - 1);
    const int w  = (tid >> 6)  & (W - 1);
    const int h  = (tid >> 12) & (H - 1);
    const int b  =  tid >> 18;

    // mask = (y*Wout + x)*C + c ; C=2^8, Wout*C=2^15
    // dy = bit15, dx = bit8 ; slot = dy*2 + dx
    const int s0 = ((m.x >> 14) & 2) | ((m.x >> 8) & 1);
    const int s1 = ((m.y >> 14) & 2) | ((m.y >> 8) & 1);
    const int s2 = ((m.z >> 14) & 2) | ((m.z >> 8) & 1);
    const int s3 = ((m.w >> 14) & 2) | ((m.w >> 8) & 1);

    v4f o[4];
#pragma unroll
    for (int s = 0; s < 4; ++s) {
        o[s].x = (s0 == s) ? v.x : 0.0f;
        o[s].y = (s1 == s) ? v.y : 0.0f;
        o[s].z = (s2 == s) ? v.z : 0.0f;
        o[s].w = (s3 == s) ? v.w : 0.0f;
    }

    // Output base (vec4 units) of window slot (dy=0,dx=0).
    const int base = ((b * Hout + 2 * h) * Wout + 2 * w) * C4 + c4;

    // Write-once, larger-than-L2 output: non-temporal B128 stores.
    // Per wave32: four 512B contiguous bursts -> HBM friendly.
    __builtin_nontemporal_store(o[0], &out[base]);
    __builtin_nontemporal_store(o[1], &out[base + C4]);
    __builtin_nontemporal_store(o[2], &out[base + Wout * C4]);
    __builtin_nontemporal_store(o[3], &out[base + Wout * C4 + C4]);
}

extern "C" void kernel_launch(void* const* d_in, const int* in_sizes, int n_in,
                              void* d_out, int out_size, void* d_ws, size_t ws_size,
                              hipStream_t stream) {
    (void)in_sizes; (void)n_in; (void)out_size; (void)d_ws; (void)ws_size;
    const v4f* upd = reinterpret_cast<const v4f*>(d_in[0]);
    const v4i* msk = reinterpret_cast<const v4i*>(d_in[1]);
    v4f*       out = reinterpret_cast<v4f*>(d_out);

    // Every output element belongs to exactly one thread's 2x2 window, so no
    // pre-zeroing pass is needed: this single kernel fully defines d_out.
    AdvancedUpSampling2D_unpool_kernel<<<TOTAL_T / BLOCK, BLOCK, 0, stream>>>(upd, msk, out);
}